// RelationalMessagePassing_84842783965196
// MI455X (gfx1250) — compile-verified
//
#include <hip/hip_runtime.h>

// RGCN-style relational message passing, fused single kernel, fp32 WMMA.
// out[b,n,h] = relu( sum_r (A[b,r] @ F[b]) @ W_r  +  F[b] @ W_0 )
//
// v2: K-pair-interleaved B panels (single ds_load_b64 per B fragment, no
// repack movs) + register-staged global prefetch one chunk ahead so global
// latency overlaps the WMMA burst of the current chunk.

typedef float v2f __attribute__((ext_vector_type(2)));
typedef float v8f __attribute__((ext_vector_type(8)));

#define BS_    8
#define NREL   16
#define NN     512   // nodes (N == M)
#define EMB    256   // D == H

#define TN     32    // node rows per workgroup tile
#define KC     32    // K chunk for all GEMM stages
#define LDA    40    // padded LDS ld for A chunk (160B rows: float4-aligned)
#define LDM    260   // padded LDS ld for msg/F-row buffer (conflict-free A-frag reads)

// ---------------- panel staging: 32 x 256 fp32, K-pair interleaved ----------
// LDS layout: element (k, c) lives at sP[(k>>1)*(2*EMB) + c*2 + (k&1)]
// -> the (k, k+1) pair for a column is a contiguous, 8B-aligned float2.

__device__ __forceinline__ void panel_global_load(const float* __restrict__ g,
                                                  float4 pa[4], float4 pb[4], int tid) {
#pragma unroll
  for (int i = 0; i < 4; ++i) {
    int id = tid + i * 256;        // task 0..1023: pair-row p, column quad q
    int p  = id >> 6;              // 0..15
    int q  = (id & 63) << 2;       // 0..252
    pa[i] = *(const float4*)(g + (size_t)(2 * p)     * EMB + q);
    pb[i] = *(const float4*)(g + (size_t)(2 * p + 1) * EMB + q);
  }
}

__device__ __forceinline__ void panel_lds_store(float* __restrict__ sP,
                                                const float4 pa[4], const float4 pb[4],
                                                int tid) {
#pragma unroll
  for (int i = 0; i < 4; ++i) {
    int id = tid + i * 256;
    int p  = id >> 6;
    int q  = (id & 63) << 2;
    float* d = sP + p * (2 * EMB) + q * 2;
    *(float2*)(d + 0) = make_float2(pa[i].x, pb[i].x);
    *(float2*)(d + 2) = make_float2(pa[i].y, pb[i].y);
    *(float2*)(d + 4) = make_float2(pa[i].z, pb[i].z);
    *(float2*)(d + 6) = make_float2(pa[i].w, pb[i].w);
  }
}

// ---------------- adjacency chunk staging: 32 x 32 fp32 ---------------------

__device__ __forceinline__ float4 achunk_global_load(const float* __restrict__ Ab,
                                                     int kc, int tid) {
  int row = tid >> 3;
  int c4  = (tid & 7) << 2;
  return *(const float4*)(Ab + (size_t)row * NN + kc + c4);
}

__device__ __forceinline__ void achunk_lds_store(float* __restrict__ As, float4 v, int tid) {
  int row = tid >> 3;
  int c4  = (tid & 7) << 2;
  *(float4*)(&As[row * LDA + c4]) = v;
}

// ---------------- one K-chunk of wave-level WMMAs ---------------------------
// acc[0..3] are four 16x16 C tiles at columns h0+{0,16,32,48}. A rows come
// from sA (row-major, leading dim ldA); B comes from a pair-interleaved panel.

__device__ __forceinline__ void mma_chunk(const float* __restrict__ sA, int ldA, int rowBase,
                                          const float* __restrict__ sP, int h0,
                                          int lr, int half, v8f acc[4]) {
#pragma unroll
  for (int k = 0; k < KC; k += 4) {
    // f32 16x4 A fragment: lanes 0-15 hold K=k..k+1, lanes 16-31 hold K=k+2..k+3
    v2f a = *(const v2f*)(sA + (rowBase + lr) * ldA + k + 2 * half);
    const int pbase = (k >> 1) + half;     // K-pair index for this half
#pragma unroll
    for (int t = 0; t < 4; ++t) {
      // f32 4x16 B fragment: contiguous (K, K+1) pair at column h0+t*16+lr
      v2f bfr = *(const v2f*)(sP + pbase * (2 * EMB) + (h0 + t * 16 + lr) * 2);
      acc[t] = __builtin_amdgcn_wmma_f32_16x16x4_f32(
          /*neg_a=*/false, a, /*neg_b=*/false, bfr,
          /*c_mod=*/(short)0, acc[t], /*reuse_a=*/false, /*reuse_b=*/false);
    }
  }
}

__global__ __launch_bounds__(256)
void rgcn_fused_wmma_kernel(const float* __restrict__ features,   // [B, NN, EMB]
                            const float* __restrict__ adjacency,  // [B, R, NN, NN]
                            const float* __restrict__ Wr,         // [R, EMB, EMB]
                            const float* __restrict__ W0,         // [EMB, EMB]
                            float* __restrict__ out) {            // [B, NN, EMB]
  __shared__ float As[TN * LDA];    //  5,120 B : 32 x KC adjacency chunk
  __shared__ float sP[KC * EMB];    // 32,768 B : pair-interleaved F / W panel
  __shared__ float msgS[TN * LDM];  // 33,280 B : 32 x 256 msg tile / F rows

  const int tid  = threadIdx.x;
  const int lane = tid & 31;
  const int wave = tid >> 5;            // 0..7
  const int lr   = lane & 15;
  const int half = lane >> 4;
  const int m0   = (wave >> 2) * 16;    // 16-row band within the 32-row tile
  const int h0   = (wave & 3) * 64;     // 4 column tiles of 16 -> 64 cols per wave

  const int n0 = blockIdx.x * TN;       // node-tile origin
  const int b  = blockIdx.y;

  const float* Fb = features + (size_t)b * NN * EMB;

  v8f acc[4] = {};                      // persistent output accumulators
  float4 pa[4], pb[4];                  // panel staging registers
  float4 aStage;                        // adjacency staging register

  for (int r = 0; r < NREL; ++r) {
    v8f msg[4] = {};
    const float* Ab = adjacency + (((size_t)b * NREL + r) * NN + n0) * NN;

    // ---- Stage A: msg_tile(32x256) = A_tile(32x512) @ F(512x256), K = 512 ----
    aStage = achunk_global_load(Ab, 0, tid);
    panel_global_load(Fb, pa, pb, tid);
    for (int kc = 0; kc < NN; kc += KC) {
      __syncthreads();                  // previous compute done -> LDS reusable
      achunk_lds_store(As, aStage, tid);
      panel_lds_store(sP, pa, pb, tid);
      if (kc + KC < NN) {               // prefetch next chunk; flies during WMMAs
        aStage = achunk_global_load(Ab, kc + KC, tid);
        panel_global_load(Fb + (size_t)(kc + KC) * EMB, pa, pb, tid);
      }
      __syncthreads();
      mma_chunk(As, LDA, m0, sP, h0, lr, half, msg);
    }

    // ---- spill msg tile to LDS (C layout: VGPR v -> rows m0+v / m0+v+8) ----
    const float* Wrb = Wr + (size_t)r * EMB * EMB;
    __syncthreads();                    // all waves done with stage-A compute
    panel_global_load(Wrb, pa, pb, tid);  // prefetch W_r chunk 0 during spill
#pragma unroll
    for (int t = 0; t < 4; ++t)
#pragma unroll
      for (int v = 0; v < 8; ++v)
        msgS[(m0 + v + 8 * half) * LDM + h0 + t * 16 + lr] = msg[t][v];

    // ---- Stage B: acc += msg_tile(32x256) @ W_r(256x256), K = 256 ----
    for (int kc = 0; kc < EMB; kc += KC) {
      __syncthreads();                  // spill visible / previous compute done
      panel_lds_store(sP, pa, pb, tid);
      if (kc + KC < EMB)
        panel_global_load(Wrb + (size_t)(kc + KC) * EMB, pa, pb, tid);
      __syncthreads();
      mma_chunk(msgS + kc, LDM, m0, sP, h0, lr, half, acc);
    }
  }

  // ---- Stage C: acc += F[b, n0:n0+32, :] @ W_0, K = 256 ----
  __syncthreads();                      // last stage-B compute done
  panel_global_load(W0, pa, pb, tid);   // prefetch W_0 chunk 0
  {
    const float* g = Fb + (size_t)n0 * EMB;
#pragma unroll
    for (int i = 0; i < 8; ++i) {
      int idx = tid + i * 256;
      int row = idx >> 6;
      int c4  = (idx & 63) << 2;
      *(float4*)(&msgS[row * LDM + c4]) = *(const float4*)(g + (size_t)row * EMB + c4);
    }
  }
  for (int kc = 0; kc < EMB; kc += KC) {
    __syncthreads();
    panel_lds_store(sP, pa, pb, tid);
    if (kc + KC < EMB)
      panel_global_load(W0 + (size_t)(kc + KC) * EMB, pa, pb, tid);
    __syncthreads();
    mma_chunk(msgS + kc, LDM, m0, sP, h0, lr, half, acc);
  }

  // ---- ReLU + coalesced store (lane index = consecutive h) ----
  float* ob = out + ((size_t)b * NN + n0) * EMB;
#pragma unroll
  for (int t = 0; t < 4; ++t)
#pragma unroll
    for (int v = 0; v < 8; ++v) {
      float x = acc[t][v];
      ob[(size_t)(m0 + v + 8 * half) * EMB + h0 + t * 16 + lr] = fmaxf(x, 0.0f);
    }
}

extern "C" void kernel_launch(void* const* d_in, const int* in_sizes, int n_in,
                              void* d_out, int out_size, void* d_ws, size_t ws_size,
                              hipStream_t stream) {
  const float* features  = (const float*)d_in[0];  // [8, 512, 256]
  const float* adjacency = (const float*)d_in[1];  // [8, 16, 512, 512]
  const float* Wr        = (const float*)d_in[2];  // [16, 256, 256]
  const float* W0        = (const float*)d_in[3];  // [256, 256]
  float* out = (float*)d_out;                      // [8, 512, 256]

  dim3 grid(NN / TN, BS_);                         // 16 x 8 = 128 workgroups
  rgcn_fused_wmma_kernel<<<grid, 256, 0, stream>>>(features, adjacency, Wr, W0, out);
}